// OSRRouter_32392643346971
// MI455X (gfx1250) — compile-verified
//
#include <hip/hip_runtime.h>
#include <math.h>

// ---------- types for WMMA ----------
typedef float v2f __attribute__((ext_vector_type(2)));
typedef float v8f __attribute__((ext_vector_type(8)));

__device__ __forceinline__ v8f wmma4(float ax, float ay, float bx, float by, v8f c) {
    v2f a; a.x = ax; a.y = ay;
    v2f b; b.x = bx; b.y = by;
    // D = A(16x4 f32) * B(4x16 f32) + C(16x16 f32)
    return __builtin_amdgcn_wmma_f32_16x16x4_f32(
        /*neg_a=*/false, a, /*neg_b=*/false, b,
        /*c_mod=*/(short)0, c, /*reuse_a=*/false, /*reuse_b=*/false);
}

#define DIMK   2048
#define NEXP   64
#define NTOK   16384
#define NWAVE  1024          // NTOK/16 waves for the big GEMM / sinkhorn rows

// ===================================================================
// K1: e_norm[r] = experts[r] / max(||experts[r]||, 1e-12)
// ===================================================================
__global__ void k1_norm_experts(const float* __restrict__ e, float* __restrict__ e_norm) {
    __shared__ float red[256];
    const int r = blockIdx.x;
    const float* row = e + (size_t)r * DIMK;
    float s = 0.f;
    for (int k = threadIdx.x; k < DIMK; k += 256) { float v = row[k]; s += v * v; }
    red[threadIdx.x] = s;
    __syncthreads();
    for (int st = 128; st > 0; st >>= 1) {
        if (threadIdx.x < st) red[threadIdx.x] += red[threadIdx.x + st];
        __syncthreads();
    }
    const float scale = 1.f / fmaxf(sqrtf(red[0]), 1e-12f);
    for (int k = threadIdx.x; k < DIMK; k += 256)
        e_norm[(size_t)r * DIMK + k] = row[k] * scale;
}

// ===================================================================
// K2: gram = e_norm @ e_norm^T  (64x64, K=2048) via f32 WMMA.
//     gram_off = relu(gram - I). Also zero-inits log_v.
//     1 block x 512 threads = 16 waves, one 16x16 tile per wave.
// ===================================================================
__global__ void k2_gram(const float* __restrict__ e_norm,
                        float* __restrict__ gram, float* __restrict__ goff,
                        float* __restrict__ go_ws, float* __restrict__ log_v) {
    const int lane  = threadIdx.x & 31;
    const int wv    = threadIdx.x >> 5;   // 0..15
    const int mt    = wv >> 2, nt = wv & 3;
    const int row16 = lane & 15, hi = lane >> 4;

    const float* ar = e_norm + (size_t)(mt * 16 + row16) * DIMK + 4 * hi;
    const float* br = e_norm + (size_t)(nt * 16 + row16) * DIMK + 4 * hi;

    v8f acc = {0.f, 0.f, 0.f, 0.f, 0.f, 0.f, 0.f, 0.f};
    #pragma unroll 4
    for (int kb = 0; kb < DIMK; kb += 8) {
        float4 a = *(const float4*)(ar + kb);
        float4 b = *(const float4*)(br + kb);
        acc = wmma4(a.x, a.y, b.x, b.y, acc);
        acc = wmma4(a.z, a.w, b.z, b.w, acc);
    }
    #pragma unroll
    for (int v = 0; v < 8; v++) {
        const int m = mt * 16 + v + 8 * hi;
        const int n = nt * 16 + row16;
        const float g = acc[v];
        gram[m * NEXP + n] = g;
        const float go = fmaxf(g - (m == n ? 1.f : 0.f), 0.f);
        goff[m * NEXP + n]  = go;
        go_ws[m * NEXP + n] = go;
    }
    if (threadIdx.x < NEXP) log_v[threadIdx.x] = 0.f;
}

// ===================================================================
// K3: main pass. Per wave: 16 tokens x all 64 experts.
//   acc = x_tile @ e_norm^T (WMMA f32, K-permuted b128 loads)
//   fused: inv_x from per-lane sum-of-squares (+shfl_xor 16)
//   sim  = acc * inv_x[m]          -> sim output + LDS
//   rep  = sim @ gram_off (LDS)    -> logK = (sim - 0.5*rep)/eps
//   logK written into probs region (overwritten in place by F).
// ===================================================================
__global__ void k3_sim_score(const float* __restrict__ x,
                             const float* __restrict__ e_norm,
                             const float* __restrict__ gram_off,
                             float* __restrict__ sim_out,
                             float* __restrict__ logK_out) {
    __shared__ float goLDS[NEXP * NEXP];        // 16 KB
    __shared__ float simLDS[8][16 * NEXP];      // 32 KB
    __shared__ float invLDS[8][16];

    for (int t = threadIdx.x; t < NEXP * NEXP; t += blockDim.x) goLDS[t] = gram_off[t];
    __syncthreads();

    const int lane  = threadIdx.x & 31;
    const int wv    = threadIdx.x >> 5;
    const int wave  = blockIdx.x * (blockDim.x >> 5) + wv;   // 0..1023
    const int row16 = lane & 15, hi = lane >> 4;
    const size_t m0 = (size_t)wave * 16;

    const float* xr = x + (m0 + row16) * DIMK + 4 * hi;
    const float* e0 = e_norm + (size_t)( 0 + row16) * DIMK + 4 * hi;
    const float* e1 = e_norm + (size_t)(16 + row16) * DIMK + 4 * hi;
    const float* e2 = e_norm + (size_t)(32 + row16) * DIMK + 4 * hi;
    const float* e3 = e_norm + (size_t)(48 + row16) * DIMK + 4 * hi;

    v8f acc0 = {0.f,0.f,0.f,0.f,0.f,0.f,0.f,0.f};
    v8f acc1 = acc0, acc2 = acc0, acc3 = acc0;
    float ssq = 0.f;

    #pragma unroll 2
    for (int kb = 0; kb < DIMK; kb += 8) {
        float4 a  = *(const float4*)(xr + kb);
        float4 b0 = *(const float4*)(e0 + kb);
        float4 b1 = *(const float4*)(e1 + kb);
        float4 b2 = *(const float4*)(e2 + kb);
        float4 b3 = *(const float4*)(e3 + kb);
        ssq += a.x*a.x + a.y*a.y + a.z*a.z + a.w*a.w;
        acc0 = wmma4(a.x, a.y, b0.x, b0.y, acc0);
        acc0 = wmma4(a.z, a.w, b0.z, b0.w, acc0);
        acc1 = wmma4(a.x, a.y, b1.x, b1.y, acc1);
        acc1 = wmma4(a.z, a.w, b1.z, b1.w, acc1);
        acc2 = wmma4(a.x, a.y, b2.x, b2.y, acc2);
        acc2 = wmma4(a.z, a.w, b2.z, b2.w, acc2);
        acc3 = wmma4(a.x, a.y, b3.x, b3.y, acc3);
        acc3 = wmma4(a.z, a.w, b3.z, b3.w, acc3);
    }

    // full-row sum of squares: lane & lane^16 cover complementary k-halves
    ssq += __shfl_xor(ssq, 16);
    if (hi == 0) invLDS[wv][row16] = 1.f / fmaxf(sqrtf(ssq), 1e-12f);
    __syncthreads();

    // sim = acc * inv_x[m]; write to sim output + LDS
    {
        v8f accs[4] = {acc0, acc1, acc2, acc3};
        #pragma unroll
        for (int t = 0; t < 4; t++) {
            #pragma unroll
            for (int v = 0; v < 8; v++) {
                const int m = v + 8 * hi;
                const int n = t * 16 + row16;
                const float s = accs[t][v] * invLDS[wv][m];
                simLDS[wv][m * NEXP + n] = s;
                sim_out[(m0 + m) * NEXP + n] = s;
            }
        }
    }
    __syncthreads();

    // repulsion + logK = (sim - 0.5*rep) / 0.05
    #pragma unroll
    for (int v = 0; v < 8; v++) {
        const int m = v + 8 * hi;
        const float* srow = &simLDS[wv][m * NEXP];
        float r0 = 0.f, r1 = 0.f, r2 = 0.f, r3 = 0.f;
        for (int j = 0; j < NEXP; j++) {
            const float sv = srow[j];
            const float* g = &goLDS[j * NEXP];
            r0 += sv * g[ 0 + row16];
            r1 += sv * g[16 + row16];
            r2 += sv * g[32 + row16];
            r3 += sv * g[48 + row16];
        }
        const size_t base = (m0 + m) * NEXP;
        logK_out[base +  0 + row16] = (srow[ 0 + row16] - 0.5f * r0) * 20.f;
        logK_out[base + 16 + row16] = (srow[16 + row16] - 0.5f * r1) * 20.f;
        logK_out[base + 32 + row16] = (srow[32 + row16] - 0.5f * r2) * 20.f;
        logK_out[base + 48 + row16] = (srow[48 + row16] - 0.5f * r3) * 20.f;
    }
}

// ===================================================================
// S1: one sinkhorn half-iteration pair start:
//   per row i: log_u[i] = -LSE_j(logK[i][j] + log_v[j])
//   streaming column-LSE partials of (logK[i][j] + log_u[i]) per wave.
// One wave handles 16 rows; lane holds columns 2*lane, 2*lane+1.
// ===================================================================
__global__ void s1_rows(const float* __restrict__ logK, const float* __restrict__ log_v,
                        float* __restrict__ log_u,
                        float* __restrict__ part_m, float* __restrict__ part_s) {
    const int lane = threadIdx.x & 31;
    const int wave = (blockIdx.x * blockDim.x + threadIdx.x) >> 5;  // 0..1023
    const float lv0 = log_v[2 * lane], lv1 = log_v[2 * lane + 1];
    float cm0 = -INFINITY, cs0 = 0.f, cm1 = -INFINITY, cs1 = 0.f;
    const size_t base = (size_t)wave * 16 * NEXP;

    for (int r = 0; r < 16; r++) {
        float2 lk = *(const float2*)(logK + base + (size_t)r * NEXP + 2 * lane);
        float a0 = lk.x + lv0, a1 = lk.y + lv1;
        float mr = fmaxf(a0, a1);
        for (int o = 16; o; o >>= 1) mr = fmaxf(mr, __shfl_xor(mr, o));
        float s = expf(a0 - mr) + expf(a1 - mr);
        for (int o = 16; o; o >>= 1) s += __shfl_xor(s, o);
        const float lu = -(mr + logf(s));
        if (lane == 0) log_u[wave * 16 + r] = lu;
        // streaming logsumexp accumulation down the columns
        const float c0 = lk.x + lu, c1 = lk.y + lu;
        if (c0 > cm0) { cs0 = cs0 * expf(cm0 - c0) + 1.f; cm0 = c0; } else cs0 += expf(c0 - cm0);
        if (c1 > cm1) { cs1 = cs1 * expf(cm1 - c1) + 1.f; cm1 = c1; } else cs1 += expf(c1 - cm1);
    }
    part_m[wave * NEXP + 2 * lane]     = cm0;
    part_s[wave * NEXP + 2 * lane]     = cs0;
    part_m[wave * NEXP + 2 * lane + 1] = cm1;
    part_s[wave * NEXP + 2 * lane + 1] = cs1;
}

// ===================================================================
// S2: finish column LSE: log_v[j] = log(B/E) - LSE over 1024 wave partials.
// One block per column.
// ===================================================================
__global__ void s2_cols(const float* __restrict__ part_m, const float* __restrict__ part_s,
                        float* __restrict__ log_v) {
    __shared__ float sm[256], ss[256];
    const int j = blockIdx.x;
    float M = -INFINITY, S = 0.f;
    for (int w = threadIdx.x; w < NWAVE; w += 256) {
        const float m2 = part_m[w * NEXP + j];
        const float s2 = part_s[w * NEXP + j];
        if (m2 > M) { S = S * expf(M - m2) + s2; M = m2; } else { S += s2 * expf(m2 - M); }
    }
    sm[threadIdx.x] = M; ss[threadIdx.x] = S;
    __syncthreads();
    for (int st = 128; st > 0; st >>= 1) {
        if (threadIdx.x < st) {
            float M1 = sm[threadIdx.x], S1 = ss[threadIdx.x];
            const float m2 = sm[threadIdx.x + st], s2 = ss[threadIdx.x + st];
            if (m2 > M1) { S1 = S1 * expf(M1 - m2) + s2; M1 = m2; } else { S1 += s2 * expf(m2 - M1); }
            sm[threadIdx.x] = M1; ss[threadIdx.x] = S1;
        }
        __syncthreads();
    }
    if (threadIdx.x == 0)
        log_v[j] = logf((float)NTOK / (float)NEXP) - (sm[0] + logf(ss[0]));
}

// ===================================================================
// F: probs = exp(logK + log_u[i] + log_v[j])  (in place over logK region)
// ===================================================================
__global__ void f_probs(float* __restrict__ p, const float* __restrict__ log_u,
                        const float* __restrict__ log_v) {
    const int t = blockIdx.x * blockDim.x + threadIdx.x;
    const int i = t >> 6, j = t & 63;
    p[t] = expf(p[t] + log_u[i] + log_v[j]);
}

// ===================================================================
extern "C" void kernel_launch(void* const* d_in, const int* in_sizes, int n_in,
                              void* d_out, int out_size, void* d_ws, size_t ws_size,
                              hipStream_t stream) {
    (void)in_sizes; (void)n_in; (void)out_size; (void)ws_size;
    const float* x       = (const float*)d_in[0];   // [16384, 2048]
    const float* experts = (const float*)d_in[1];   // [64, 2048]

    float* out   = (float*)d_out;
    float* probs = out;                       // [16384*64]  (holds logK until F)
    float* sim   = out + (size_t)NTOK * NEXP; // [16384*64]
    float* gram  = sim + (size_t)NTOK * NEXP; // [64*64]
    float* goff  = gram + NEXP * NEXP;        // [64*64]

    float* ws     = (float*)d_ws;
    float* e_norm = ws;                                   // 64*2048
    float* go_ws  = e_norm + (size_t)NEXP * DIMK;         // 64*64
    float* log_u  = go_ws + NEXP * NEXP;                  // 16384
    float* log_v  = log_u + NTOK;                         // 64
    float* part_m = log_v + NEXP;                         // 1024*64
    float* part_s = part_m + (size_t)NWAVE * NEXP;        // 1024*64

    k1_norm_experts<<<NEXP, 256, 0, stream>>>(experts, e_norm);
    k2_gram<<<1, 512, 0, stream>>>(e_norm, gram, goff, go_ws, log_v);
    k3_sim_score<<<NWAVE / 8, 256, 0, stream>>>(x, e_norm, go_ws, sim, probs);
    for (int it = 0; it < 5; it++) {
        s1_rows<<<NWAVE / 8, 256, 0, stream>>>(probs, log_v, log_u, part_m, part_s);
        s2_cols<<<NEXP, 256, 0, stream>>>(part_m, part_s, log_v);
    }
    f_probs<<<(NTOK * NEXP) / 256, 256, 0, stream>>>(probs, log_u, log_v);
}